// PDN_dense_add_skip_1bro_pre_batch_act_max_test_38010460569941
// MI455X (gfx1250) — compile-verified
//
#include <hip/hip_runtime.h>
#include <hip/hip_bf16.h>
#include <math.h>

#define Nn     100000
#define Ee     1600000
#define Gg     2048
#define FIN    64
#define Cc     128
#define EDIM   16
#define LL     4
#define NLAYER 5          // conv1 + 4 hidden convs
#define EPSf   1e-5f

typedef __attribute__((ext_vector_type(2))) float v2f;
typedef __attribute__((ext_vector_type(8))) float v8f;

// ---------------------------------------------------------------- utilities
__global__ void fillf_kernel(float* __restrict__ p, float v, int n) {
  int i = blockIdx.x * blockDim.x + threadIdx.x;
  if (i < n) p[i] = v;
}

// ------------------------------------------------- edge gate MLP (all 5 layers)
// w_l[e] = sigmoid( relu(edge_attr[e] @ mW1_l + mb1_l) @ mW2_l + mb2_l )
__global__ void edge_mlp_kernel(const float* __restrict__ ea,
                                const float* __restrict__ c_mW1, const float* __restrict__ c_mb1,
                                const float* __restrict__ c_mW2, const float* __restrict__ c_mb2,
                                const float* __restrict__ h_mW1, const float* __restrict__ h_mb1,
                                const float* __restrict__ h_mW2, const float* __restrict__ h_mb2,
                                float* __restrict__ gates) {
  __shared__ float sW1[NLAYER * 256];
  __shared__ float sb1[NLAYER * 16];
  __shared__ float sW2[NLAYER * 16];
  __shared__ float sb2[NLAYER];
  for (int i = threadIdx.x; i < NLAYER * 256; i += blockDim.x) {
    int l = i >> 8, r = i & 255;
    sW1[i] = (l == 0) ? c_mW1[r] : h_mW1[(l - 1) * 256 + r];
  }
  for (int i = threadIdx.x; i < NLAYER * 16; i += blockDim.x) {
    int l = i >> 4, r = i & 15;
    sb1[i] = (l == 0) ? c_mb1[r] : h_mb1[(l - 1) * 16 + r];
    sW2[i] = (l == 0) ? c_mW2[r] : h_mW2[(l - 1) * 16 + r];
  }
  if (threadIdx.x < NLAYER)
    sb2[threadIdx.x] = (threadIdx.x == 0) ? c_mb2[0] : h_mb2[threadIdx.x - 1];
  __syncthreads();

  int e = blockIdx.x * blockDim.x + threadIdx.x;
  if (e >= Ee) return;

  float x[EDIM];
  const float4* row = (const float4*)(ea + (size_t)e * EDIM);
  float4 r0 = row[0], r1 = row[1], r2 = row[2], r3 = row[3];
  x[0]=r0.x; x[1]=r0.y; x[2]=r0.z; x[3]=r0.w;
  x[4]=r1.x; x[5]=r1.y; x[6]=r1.z; x[7]=r1.w;
  x[8]=r2.x; x[9]=r2.y; x[10]=r2.z; x[11]=r2.w;
  x[12]=r3.x; x[13]=r3.y; x[14]=r3.z; x[15]=r3.w;

  for (int l = 0; l < NLAYER; ++l) {
    const float* W1 = &sW1[l * 256];
    float s = sb2[l];
    #pragma unroll
    for (int j = 0; j < EDIM; ++j) {
      float hj = sb1[l * 16 + j];
      #pragma unroll
      for (int i = 0; i < EDIM; ++i)
        hj = fmaf(x[i], W1[i * 16 + j], hj);
      hj = fmaxf(hj, 0.0f);
      s = fmaf(hj, sW2[l * 16 + j], s);
    }
    gates[(size_t)l * Ee + e] = 1.0f / (1.0f + expf(-s));
  }
}

// ------------------------------------------------- degree accumulation (deg init = 1.0 self-loop)
__global__ void deg_acc_kernel(const int* __restrict__ ei,
                               const float* __restrict__ gates,
                               float* __restrict__ deg) {
  int e = blockIdx.x * blockDim.x + threadIdx.x;
  if (e >= Ee) return;
  int dst = ei[Ee + e];
  #pragma unroll
  for (int l = 0; l < NLAYER; ++l)
    atomicAdd(&deg[(size_t)l * Nn + dst], gates[(size_t)l * Ee + e]);
}

__global__ void deg_to_dinv_kernel(float* __restrict__ deg, int n) {
  int i = blockIdx.x * blockDim.x + threadIdx.x;
  if (i < n) deg[i] = rsqrtf(deg[i]);   // deg >= 1 always (self-loop weight 1)
}

// ------------------------------------------------- fused conv GEMM (WMMA f32)
// T = act(A) @ W, h = bias + T * dinv^2, where act = BN+ReLU (use_bn) or identity.
// A: [Nn,K] row-major; W: [K,128] row-major. Block = 16 rows x 128 cols = 8 waves,
// each wave one 16x16 tile via V_WMMA_F32_16X16X4_F32. The 16xK A-tile is staged
// once per block in LDS (row stride K+4 -> bank-conflict-free ds reads) and the
// activation is applied during staging. In-place A==h is safe: a block reads only
// the 16 rows it later writes, separated by __syncthreads().
__global__ void gemm_conv_fused_kernel(const float* __restrict__ A,
                                       const float* __restrict__ W,
                                       const float* __restrict__ stats, int use_bn,
                                       const float* __restrict__ dinv,
                                       const float* __restrict__ bias,
                                       float* __restrict__ T,
                                       float* __restrict__ h,
                                       int K) {
  __shared__ float sA[16 * (Cc + 4)];
  const int stride = K + 4;
  const int row0   = blockIdx.x << 4;

  // ---- stage A tile with fused BN+ReLU (K divides 256 -> channel fixed per thread)
  {
    const int c = threadIdx.x % K;
    float mu = 0.0f, rs = 1.0f;
    if (use_bn) {
      mu = stats[c] * (1.0f / Nn);
      float var = stats[128 + c] * (1.0f / Nn) - mu * mu;
      rs = rsqrtf(var + EPSf);
    }
    for (int idx = threadIdx.x; idx < 16 * K; idx += 256) {
      int r = idx / K;
      float v = A[(size_t)(row0 + r) * K + c];
      if (use_bn) v = fmaxf((v - mu) * rs, 0.0f);
      sA[r * stride + c] = v;
    }
  }
  __syncthreads();

  const int lane  = threadIdx.x & 31;
  const int wave  = threadIdx.x >> 5;
  const int col0  = wave << 4;
  const int laneM = lane & 15;
  const int khalf = (lane >> 4) << 1;   // lanes 16..31 hold K+2,K+3 halves

  v8f acc = {};
  const float* arow = &sA[laneM * stride];
  for (int k = 0; k < K; k += 4) {
    const int kk = k + khalf;
    v2f a, b;
    a.x = arow[kk];
    a.y = arow[kk + 1];
    b.x = W[(size_t)kk * Cc + col0 + laneM];
    b.y = W[(size_t)(kk + 1) * Cc + col0 + laneM];
    acc = __builtin_amdgcn_wmma_f32_16x16x4_f32(false, a, false, b,
                                                (short)0, acc, false, false);
  }

  // D layout: VGPR v -> M = v (lanes 0-15) / v+8 (lanes 16-31), N = lane&15.
  // Epilogue writes T and the self-loop/bias-initialized conv output h.
  const int   mbase = (lane >> 4) << 3;
  const float bcol  = bias[col0 + laneM];
  #pragma unroll
  for (int v = 0; v < 8; ++v) {
    const int rowv = row0 + mbase + v;
    const size_t off = (size_t)rowv * Cc + col0 + laneM;
    const float t = acc[v];
    const float di = dinv[rowv];
    T[off] = t;
    h[off] = bcol + t * di * di;      // self-loop norm = dinv^2, weight 1
  }
}

// ------------------------------------------------- edge scatter: h[dst] += T[src] * norm(e)
// warp per edge: 32 lanes x float4 = 128 channels
__global__ void scatter_edges_kernel(const int* __restrict__ ei,
                                     const float* __restrict__ gates,
                                     const float* __restrict__ dinv,
                                     const float* __restrict__ T,
                                     float* __restrict__ h) {
  int warp = (blockIdx.x * blockDim.x + threadIdx.x) >> 5;
  int lane = threadIdx.x & 31;
  if (warp >= Ee) return;
  int src = ei[warp];
  int dst = ei[Ee + warp];
  float nrm = dinv[src] * gates[warp] * dinv[dst];
  const float4 t4 = *(const float4*)(T + (size_t)src * Cc + lane * 4);
  float* hp = h + (size_t)dst * Cc + lane * 4;
  atomicAdd(hp + 0, t4.x * nrm);
  atomicAdd(hp + 1, t4.y * nrm);
  atomicAdd(hp + 2, t4.z * nrm);
  atomicAdd(hp + 3, t4.w * nrm);
}

// ------------------------------------------------- batch-norm statistics (sum, sumsq per channel)
__global__ void bn_stats_kernel(const float* __restrict__ h, float* __restrict__ stats) {
  __shared__ float ls[256], lq[256];
  const int t = threadIdx.x;
  const int c = t & 127;
  float s = 0.f, q = 0.f;
  for (int n = blockIdx.x * 2 + (t >> 7); n < Nn; n += gridDim.x * 2) {
    float v = h[(size_t)n * Cc + c];
    s += v; q += v * v;
  }
  ls[t] = s; lq[t] = q;
  __syncthreads();
  if (t < 128) {
    atomicAdd(&stats[c],       ls[t] + ls[t + 128]);
    atomicAdd(&stats[128 + c], lq[t] + lq[t + 128]);
  }
}

// ------------------------------------------------- residual relu + segment_max pooling
// values are >= 0 after relu, so int-bit atomicMax is order-correct and init 0 is safe
__global__ void pool_max_kernel(const float* __restrict__ h,
                                const float* __restrict__ h0,
                                const int* __restrict__ batch,
                                float* __restrict__ pooled) {
  int idx = blockIdx.x * blockDim.x + threadIdx.x;   // over Nn*Cc
  int n = idx >> 7, c = idx & 127;
  float v = fmaxf(h[idx] + h0[idx], 0.0f);
  atomicMax((int*)(pooled + (size_t)batch[n] * Cc + c), __float_as_int(v));
}

// ------------------------------------------------- final linear [G,128]@[128,2]
__global__ void final_linear_kernel(const float* __restrict__ pooled,
                                    const float* __restrict__ W,
                                    const float* __restrict__ b,
                                    float* __restrict__ out) {
  int idx = blockIdx.x * blockDim.x + threadIdx.x;
  if (idx >= Gg * 2) return;
  int g = idx >> 1, k = idx & 1;
  const float* p = pooled + (size_t)g * Cc;
  float s = b[k];
  #pragma unroll 4
  for (int c = 0; c < Cc; ++c)
    s = fmaf(p[c], W[c * 2 + k], s);
  out[idx] = s;
}

// ================================================================ launch
extern "C" void kernel_launch(void* const* d_in, const int* in_sizes, int n_in,
                              void* d_out, int out_size, void* d_ws, size_t ws_size,
                              hipStream_t stream) {
  const float* x       = (const float*)d_in[0];
  const float* eattr   = (const float*)d_in[1];
  const float* conv1_W = (const float*)d_in[2];
  const float* conv1_b = (const float*)d_in[3];
  const float* c_mW1   = (const float*)d_in[4];
  const float* c_mb1   = (const float*)d_in[5];
  const float* c_mW2   = (const float*)d_in[6];
  const float* c_mb2   = (const float*)d_in[7];
  const float* hW      = (const float*)d_in[8];
  const float* hb      = (const float*)d_in[9];
  const float* hmW1    = (const float*)d_in[10];
  const float* hmb1    = (const float*)d_in[11];
  const float* hmW2    = (const float*)d_in[12];
  const float* hmb2    = (const float*)d_in[13];
  const float* lin_W   = (const float*)d_in[14];
  const float* lin_b   = (const float*)d_in[15];
  const int*   eidx    = (const int*)d_in[16];
  const int*   batch   = (const int*)d_in[17];
  float* out = (float*)d_out;

  // workspace carving (~190 MB of floats)
  float* ws     = (float*)d_ws;
  float* gates  = ws;  ws += (size_t)NLAYER * Ee;   // 8,000,000
  float* dinv   = ws;  ws += (size_t)NLAYER * Nn;   //   500,000 (deg -> dinv in place)
  float* T      = ws;  ws += (size_t)Nn * Cc;       // 12,800,000
  float* h0     = ws;  ws += (size_t)Nn * Cc;
  float* h      = ws;  ws += (size_t)Nn * Cc;
  float* stats  = ws;  ws += 256;
  float* pooled = ws;  ws += (size_t)Gg * Cc;

  const int NT = 256;
  const int NC = Nn * Cc;                           // 12,800,000

  // init: self-loop degrees and pooled
  fillf_kernel<<<(NLAYER * Nn + NT - 1) / NT, NT, 0, stream>>>(dinv, 1.0f, NLAYER * Nn);
  fillf_kernel<<<(Gg * Cc + NT - 1) / NT, NT, 0, stream>>>(pooled, 0.0f, Gg * Cc);

  // all 5 layers of edge gates in one pass over edge_attr
  edge_mlp_kernel<<<(Ee + NT - 1) / NT, NT, 0, stream>>>(
      eattr, c_mW1, c_mb1, c_mW2, c_mb2, hmW1, hmb1, hmW2, hmb2, gates);

  // degrees and D^{-1/2}
  deg_acc_kernel<<<(Ee + NT - 1) / NT, NT, 0, stream>>>(eidx, gates, dinv);
  deg_to_dinv_kernel<<<(NLAYER * Nn + NT - 1) / NT, NT, 0, stream>>>(dinv, NLAYER * Nn);

  // conv1 -> h0 (GEMM + self-loop/bias init fused), then edge scatter
  gemm_conv_fused_kernel<<<Nn / 16, NT, 0, stream>>>(
      x, conv1_W, nullptr, 0, dinv, conv1_b, T, h0, FIN);
  scatter_edges_kernel<<<(Ee * 32 + NT - 1) / NT, NT, 0, stream>>>(
      eidx, gates, dinv, T, h0);

  // 4 hidden convs: bn stats -> fused (bn+relu+gemm+init) -> scatter
  const float* cur = h0;
  for (int l = 0; l < LL; ++l) {
    fillf_kernel<<<1, 256, 0, stream>>>(stats, 0.0f, 256);
    bn_stats_kernel<<<256, NT, 0, stream>>>(cur, stats);
    gemm_conv_fused_kernel<<<Nn / 16, NT, 0, stream>>>(
        cur, hW + (size_t)l * Cc * Cc, stats, 1,
        dinv + (size_t)(l + 1) * Nn, hb + (size_t)l * Cc, T, h, Cc);
    scatter_edges_kernel<<<(Ee * 32 + NT - 1) / NT, NT, 0, stream>>>(
        eidx, gates + (size_t)(l + 1) * Ee, dinv + (size_t)(l + 1) * Nn, T, h);
    cur = h;
  }

  // residual relu + segment_max pool + final linear
  pool_max_kernel<<<(NC + NT - 1) / NT, NT, 0, stream>>>(h, h0, batch, pooled);
  final_linear_kernel<<<(Gg * 2 + NT - 1) / NT, NT, 0, stream>>>(pooled, lin_W, lin_b, out);
}